// GNN_4896262717753
// MI455X (gfx1250) — compile-verified
//
#include <hip/hip_runtime.h>

typedef float v2f __attribute__((ext_vector_type(2)));
typedef float v8f __attribute__((ext_vector_type(8)));

// ---------------------------------------------------------------------------
// Dense GEMM  H[b][n][o] = sum_k X[b][n][k] * W[o][k]   via V_WMMA_F32_16X16X4_F32
// One wave computes a 16-node x 16-feature tile. Main K-loop is guard-free:
//  - node rows always valid (N % 16 == 0)
//  - out-of-range B columns are merely clamped (they only affect C columns
//    that are discarded at store time)
//  - K tail (din % 4) handled by one masked iteration with cndmask selects
// ---------------------------------------------------------------------------
__global__ void __launch_bounds__(32)
gemm_xwT_wmma(const float* __restrict__ X, long xBatchStride, int ldx,
              const float* __restrict__ W, int din, int dout,
              float* __restrict__ H, int N)
{
    const int lane = threadIdx.x;        // 0..31
    const int half = lane >> 4;          // 0 or 1
    const int nl   = lane & 15;
    const int nodeTile = blockIdx.x * 16;
    const int outTile  = blockIdx.y * 16;
    const int b        = blockIdx.z;

    // A: lane holds row (nodeTile+nl), K = k0 + 2*half + v
    const float* Xr = X + (long)b * xBatchStride + (long)(nodeTile + nl) * ldx;
    // B: lane holds column (outTile+nl) of W^T == row of W; clamp row in-bounds
    const int bColC = (outTile + nl < dout) ? (outTile + nl) : (dout - 1);
    const float* Wr = W + (long)bColC * din;

    v8f acc = {};
    const int kMain = din & ~3;
    int k = 0;
#pragma unroll 4
    for (; k < kMain; k += 4) {
        const int ka = k + half * 2;
        v2f a, bm;
        a.x  = Xr[ka];
        a.y  = Xr[ka + 1];
        bm.x = Wr[ka];
        bm.y = Wr[ka + 1];
        acc = __builtin_amdgcn_wmma_f32_16x16x4_f32(
                  false, a, false, bm, (short)0, acc, false, false);
    }
    if (kMain < din) {                    // uniform branch (din is uniform)
        const int ka0 = k + half * 2;
        const int ka1 = ka0 + 1;
        const int kc0 = (ka0 < din) ? ka0 : (din - 1);   // clamped safe addresses
        const int kc1 = (ka1 < din) ? ka1 : (din - 1);
        const float x0 = Xr[kc0], x1 = Xr[kc1];          // unconditional loads
        const float w0 = Wr[kc0], w1 = Wr[kc1];
        v2f a, bm;
        a.x  = (ka0 < din) ? x0 : 0.0f;                  // value selects, no exec
        a.y  = (ka1 < din) ? x1 : 0.0f;
        bm.x = (ka0 < din) ? w0 : 0.0f;
        bm.y = (ka1 < din) ? w1 : 0.0f;
        acc = __builtin_amdgcn_wmma_f32_16x16x4_f32(
                  false, a, false, bm, (short)0, acc, false, false);
    }

    // C/D layout: VGPR r -> M = r + 8*half, N = lane&15
    const int f = outTile + nl;
    if (f < dout) {
        float* Hr = H + ((long)b * N + nodeTile + 8 * half) * dout + f;
#pragma unroll
        for (int r = 0; r < 8; ++r)
            Hr[(long)r * dout] = acc[r];
    }
}

// ---------------------------------------------------------------------------
// Degree / normalization helpers
// ---------------------------------------------------------------------------
__global__ void init_deg(float* __restrict__ deg, int total)
{
    int i = blockIdx.x * blockDim.x + threadIdx.x;
    if (i < total) deg[i] = 1.0f;   // self-loop weight baked in
}

__global__ void accum_deg(const float* __restrict__ caps, const int* __restrict__ col,
                          float* __restrict__ deg, int E, int N, int BE)
{
    int i = blockIdx.x * blockDim.x + threadIdx.x;
    if (i >= BE) return;
    int b = i / E, e = i - b * E;
    atomicAdd(&deg[b * N + col[e]], caps[i]);
}

__global__ void finalize_dinv(float* __restrict__ deg, int total)
{
    int i = blockIdx.x * blockDim.x + threadIdx.x;
    if (i >= total) return;
    float d = deg[i];
    deg[i] = (d > 0.0f) ? rsqrtf(d) : 0.0f;
}

// agg[b][n][f] = dinv[b][n]^2 * h[b][n][f]     (self-loop term, also init)
__global__ void selfloop_init(const float* __restrict__ hW, const float* __restrict__ dinv,
                              float* __restrict__ agg, int dout, int total)
{
    int i = blockIdx.x * blockDim.x + threadIdx.x;
    if (i >= total) return;
    int bn = i / dout;
    float d = dinv[bn];
    agg[i] = d * d * hW[i];
}

// agg[b][col][f] += dinv[row]*cap*dinv[col] * h[b][row][f]
// 4 threads per edge, each covers a quarter of the feature range.
__global__ void edge_scatter(const float* __restrict__ hW, const float* __restrict__ caps,
                             const int* __restrict__ row, const int* __restrict__ col,
                             const float* __restrict__ dinv, float* __restrict__ agg,
                             int N, int E, int dout, int total)
{
    int i = blockIdx.x * blockDim.x + threadIdx.x;
    if (i >= total) return;
    int sub = i & 3;
    int be  = i >> 2;
    int b = be / E, e = be - b * E;
    int r = row[e], c = col[e];
    float nrm = dinv[b * N + r] * caps[be] * dinv[b * N + c];
    int chunk = (dout + 3) >> 2;
    int f0 = sub * chunk;
    int f1 = f0 + chunk; if (f1 > dout) f1 = dout;
    const float* src = hW + ((long)b * N + r) * dout;
    float*       dst = agg + ((long)b * N + c) * dout;
    for (int f = f0; f < f1; ++f)
        atomicAdd(&dst[f], nrm * src[f]);
}

// ne[b][n][off+f] = leaky_relu(agg[b][n][f] + bias[f], 0.02)
__global__ void bias_leaky(const float* __restrict__ agg, const float* __restrict__ bias,
                           float* __restrict__ ne, int dout, int neStride, int neOff, int total)
{
    int i = blockIdx.x * blockDim.x + threadIdx.x;
    if (i >= total) return;
    int bn = i / dout;
    int f  = i - bn * dout;
    float v = agg[i] + bias[f];
    v = (v >= 0.0f) ? v : 0.02f * v;
    ne[(long)bn * neStride + neOff + f] = v;
}

// out[b][e][f] = ne[b][row][f] + ne[b][col][f]  (f<197) ; out[b][e][197] = caps[b][e]
__global__ void edge_out(const float* __restrict__ ne, const float* __restrict__ caps,
                         const int* __restrict__ row, const int* __restrict__ col,
                         float* __restrict__ out, int N, int E, long total)
{
    long i = (long)blockIdx.x * blockDim.x + threadIdx.x;
    if (i >= total) return;
    const int F = 198, NW = 197;
    int  f  = (int)(i % F);
    long be = i / F;
    int  b  = (int)(be / E);
    int  e  = (int)(be - (long)b * E);
    float v;
    if (f == NW) {
        v = caps[(long)b * E + e];
    } else {
        int r = row[e], c = col[e];
        v = ne[((long)b * N + r) * NW + f] + ne[((long)b * N + c) * NW + f];
    }
    out[i] = v;
}

// ---------------------------------------------------------------------------
extern "C" void kernel_launch(void* const* d_in, const int* in_sizes, int n_in,
                              void* d_out, int out_size, void* d_ws, size_t ws_size,
                              hipStream_t stream)
{
    (void)in_sizes; (void)n_in; (void)out_size; (void)ws_size;
    const int B = 4, N = 10000, E = 160000;

    const float* x0   = (const float*)d_in[0];   // [B,N,64]
    const float* caps = (const float*)d_in[1];   // [B,E]
    const int*   ei   = (const int*)  d_in[2];   // [2,E]
    const float* Wp[3] = { (const float*)d_in[3], (const float*)d_in[5], (const float*)d_in[7] };
    const float* bp[3] = { (const float*)d_in[4], (const float*)d_in[6], (const float*)d_in[8] };
    const int* row = ei;
    const int* col = ei + E;

    // workspace partition (floats)
    float* ws   = (float*)d_ws;
    float* dinv = ws;  ws += (size_t)B * N;          // deg then dinv
    float* hW   = ws;  ws += (size_t)B * N * 66;     // GEMM result
    float* agg  = ws;  ws += (size_t)B * N * 66;     // aggregation accumulator
    float* ne   = ws;  /* B*N*197 */                 // concatenated embeddings

    const int BLK = 256;
    const int totBN = B * N;
    const int BE    = B * E;

    // degrees -> dinv
    init_deg<<<(totBN + BLK - 1) / BLK, BLK, 0, stream>>>(dinv, totBN);
    accum_deg<<<(BE + BLK - 1) / BLK, BLK, 0, stream>>>(caps, col, dinv, E, N, BE);
    finalize_dinv<<<(totBN + BLK - 1) / BLK, BLK, 0, stream>>>(dinv, totBN);

    const int din_[3]  = { 64, 65, 66 };
    const int dout_[3] = { 65, 66, 66 };
    const int off_[3]  = { 0, 65, 131 };

    for (int l = 0; l < 3; ++l) {
        const float* X; long xbs; int ldx;
        if (l == 0) { X = x0;              ldx = 64;  xbs = (long)N * 64;  }
        else        { X = ne + off_[l-1];  ldx = 197; xbs = (long)N * 197; }

        dim3 g(N / 16, (dout_[l] + 15) / 16, B);
        gemm_xwT_wmma<<<g, 32, 0, stream>>>(X, xbs, ldx, Wp[l], din_[l], dout_[l], hW, N);

        int tot = B * N * dout_[l];
        selfloop_init<<<(tot + BLK - 1) / BLK, BLK, 0, stream>>>(hW, dinv, agg, dout_[l], tot);
        int scat = BE * 4;
        edge_scatter<<<(scat + BLK - 1) / BLK, BLK, 0, stream>>>(hW, caps, row, col, dinv, agg,
                                                                 N, E, dout_[l], scat);
        bias_leaky<<<(tot + BLK - 1) / BLK, BLK, 0, stream>>>(agg, bp[l], ne, dout_[l],
                                                              197, off_[l], tot);
    }

    long totOut = (long)B * E * 198;
    unsigned blocks = (unsigned)((totOut + BLK - 1) / BLK);
    edge_out<<<blocks, BLK, 0, stream>>>(ne, caps, row, col, (float*)d_out, N, E, totOut);
}